// STFT_76450417869070
// MI455X (gfx1250) — compile-verified
//
#include <hip/hip_runtime.h>

typedef __attribute__((ext_vector_type(2))) float v2f;
typedef __attribute__((ext_vector_type(8))) float v8f;

#define NFFT   2048
#define HOP    512
#define NFREQ  1025                      // K = NFFT/2 + 1
#define LSIG   262144
#define NFRM   513                       // T
#define NBC    32                        // B*C
#define FPW    32                        // frames per workgroup (two 16-frame tiles)
#define CHUNK  ((FPW - 1) * HOP + NFFT)  // 17920 staged samples
#define ROWPAD 4                         // pad words per 512-sample row (bank decorrelation)
#define LDSW   (CHUNK + ROWPAD * (CHUNK / 512) + ROWPAD)   // padded LDS words
#define IMAG_OFF ((size_t)16 * 2050 * 513)                 // flat offset of imag block in d_out

#if defined(__HIP_DEVICE_COMPILE__) && !__has_builtin(__builtin_amdgcn_wmma_f32_16x16x4_f32)
#error "missing __builtin_amdgcn_wmma_f32_16x16x4_f32 on this toolchain"
#endif

__device__ __forceinline__ v8f wmma4_f32(v2f a, v2f b, v8f c) {
#if defined(__HIP_DEVICE_COMPILE__)
  // 8 args: (neg_a, A, neg_b, B, c_mod, C, reuse_a, reuse_b)
  return __builtin_amdgcn_wmma_f32_16x16x4_f32(false, a, false, b, (short)0, c,
                                               false, false);
#else
  (void)a; (void)b;
  return c;
#endif
}

__global__ __launch_bounds__(256, 1)
void stft_wmma_f32(const float* __restrict__ x,
                   const float* __restrict__ wr,
                   const float* __restrict__ wi,
                   float* __restrict__ out) {
  __shared__ float lds[LDSW];

  const int tid = threadIdx.x;
  const int bc  = blockIdx.y;             // b*2 + c, 0..31
  const int T0  = blockIdx.x * FPW;       // first frame of this tile

  // ---- stage reflect-padded signal window into LDS (shared by 32 frames) ----
  const float* xin = x + (size_t)bc * LSIG;
  const int S0 = T0 * HOP;                // window start in padded-signal coords
  for (int i = tid; i < CHUNK; i += 256) {          // 17920 = 70*256, no tail
    int j = S0 + i - (NFFT / 2);                    // reflect: pad = 1024
    j = (j < 0) ? -j : j;
    j = (j >= LSIG) ? (2 * LSIG - 2 - j) : j;
    lds[i + ROWPAD * (i >> 9)] = xin[j];
  }
  __syncthreads();

  const int lane  = tid & 31;
  const int wave  = tid >> 5;
  const int lrow  = lane & 15;            // row-within-tile (M for A, N for B)
  const int khalf = lane >> 4;            // 0: K=0,1   1: K=2,3 (ISA 16x4 layout)

  // B fragment (frames) LDS byte offsets; padded row stride = 516 words
  const int rowstride = HOP + ROWPAD;     // 516
  const int b_off0 = (lrow * rowstride + 2 * khalf) * 4;           // frame tile 0
  const int b_off1 = ((lrow + 16) * rowstride + 2 * khalf) * 4;    // frame tile 1

  float* oR = out + (size_t)NFREQ * NFRM * bc;      // real block, this (b,c)
  float* oI = oR + IMAG_OFF;                        // imag block, same layout

  // ---- each wave: columns of 16 freq bins, real+imag, two frame tiles ----
  for (int col = wave; col < (NFREQ + 15) / 16; col += 8) {
    const int fb   = col * 16;
    int rowc = fb + lrow;
    rowc = (rowc > NFREQ - 1) ? (NFREQ - 1) : rowc; // clamp ragged last column
    const float* ar = wr + (size_t)rowc * NFFT + 2 * khalf;
    const float* ai = wi + (size_t)rowc * NFFT + 2 * khalf;

    v8f cR0 = {}, cI0 = {}, cR1 = {}, cI1 = {};

    for (int kb = 0; kb < NFFT / 512; ++kb) {       // 4 blocks of 512 samples
      const int kbase = kb * 512;
      const char* ldsb = (const char*)lds + kb * rowstride * 4;
#pragma unroll 4
      for (int kk = 0; kk < 512; kk += 4) {
        v2f a_r = *(const v2f*)(ar + kbase + kk);               // weights (L2)
        v2f a_i = *(const v2f*)(ai + kbase + kk);
        v2f b0  = *(const v2f*)(ldsb + b_off0 + kk * 4);        // frames (LDS)
        v2f b1  = *(const v2f*)(ldsb + b_off1 + kk * 4);
        cR0 = wmma4_f32(a_r, b0, cR0);
        cI0 = wmma4_f32(a_i, b0, cI0);
        cR1 = wmma4_f32(a_r, b1, cR1);
        cI1 = wmma4_f32(a_i, b1, cI1);
      }
    }

    // ---- store: lane -> t (contiguous), VGPR r -> freq row (r + 8*khalf) ----
    const int t0 = T0 + lrow;
    const int t1 = t0 + 16;
#pragma unroll
    for (int r = 0; r < 8; ++r) {
      const int f = fb + r + 8 * khalf;
      if (f < NFREQ) {
        const size_t base = (size_t)f * NFRM;
        if (t0 < NFRM) { oR[base + t0] = cR0[r]; oI[base + t0] = cI0[r]; }
        if (t1 < NFRM) { oR[base + t1] = cR1[r]; oI[base + t1] = cI1[r]; }
      }
    }
  }
}

extern "C" void kernel_launch(void* const* d_in, const int* in_sizes, int n_in,
                              void* d_out, int out_size, void* d_ws, size_t ws_size,
                              hipStream_t stream) {
  (void)in_sizes; (void)n_in; (void)d_ws; (void)ws_size; (void)out_size;
  const float* x  = (const float*)d_in[0];   // (16, 2, 262144) f32
  const float* wr = (const float*)d_in[1];   // (1025, 2048)    f32
  const float* wi = (const float*)d_in[2];   // (1025, 2048)    f32
  float* out = (float*)d_out;                // real || imag, each (16, 2050, 513)

  dim3 grid((NFRM + FPW - 1) / FPW, NBC);    // 17 x 32 workgroups
  stft_wmma_f32<<<grid, 256, 0, stream>>>(x, wr, wi, out);
}